// MBGCN_18296560681450
// MI455X (gfx1250) — compile-verified
//
#include <hip/hip_runtime.h>
#include <hip/hip_bf16.h>
#include <cstdint>

#define NUM_USERS 100000
#define NUM_ITEMS 50000
#define EMBED     64
#define RREL      3
#define NNZ_E     1000000
#define IG_NNZ_E  500000
#define BATCH     512
#define KITEMS    100
#define EPSF      1e-8f
#define LAMBF     0.5f
#define L2NORMF   1e-4f

typedef __attribute__((ext_vector_type(2))) float v2f;
typedef __attribute__((ext_vector_type(8))) float v8f;
typedef __attribute__((address_space(3))) float lds_float;

// ---------------------------------------------------------------------------
// user -> batch slot map (last writer wins; duplicates fixed up later)
__global__ void k_slot(const int* __restrict__ user, int* __restrict__ u_slot) {
    int b = blockIdx.x * blockDim.x + threadIdx.x;
    if (b < BATCH) u_slot[user[b]] = b;
}

// ub_deg[r][u] = # rel edges with row u (relation r)
__global__ void k_ubdeg(const int* __restrict__ rel_rows, float* __restrict__ ub_deg) {
    int idx = blockIdx.x * blockDim.x + threadIdx.x;
    if (idx >= RREL * NNZ_E) return;
    int r = idx / NNZ_E;
    atomicAdd(&ub_deg[(size_t)r * NUM_USERS + rel_rows[idx]], 1.0f);
}

// ig_deg[i] = sum of vals over rows (single relation; ptrs pre-offset)
__global__ void k_igdeg(const int* __restrict__ rows, const float* __restrict__ vals,
                        float* __restrict__ ig_deg) {
    int e = blockIdx.x * blockDim.x + threadIdx.x;
    if (e >= IG_NNZ_E) return;
    atomicAdd(&ig_deg[rows[e]], vals[e]);
}

// agg[row] += val * item_embedding[col]   (64 lanes per edge)
__global__ void k_agg(const int* __restrict__ rows, const int* __restrict__ cols,
                      const float* __restrict__ vals, const float* __restrict__ ie,
                      float* __restrict__ agg) {
    int idx = blockIdx.x * blockDim.x + threadIdx.x;
    if (idx >= IG_NNZ_E * EMBED) return;
    int e = idx >> 6, j = idx & 63;
    atomicAdd(&agg[(size_t)rows[e] * EMBED + j], vals[e] * ie[(size_t)cols[e] * EMBED + j]);
}

__global__ void k_norm_agg(const float* __restrict__ ig_deg, float* __restrict__ agg) {
    int idx = blockIdx.x * blockDim.x + threadIdx.x;
    if (idx >= NUM_ITEMS * EMBED) return;
    agg[idx] /= (ig_deg[idx >> 6] + EPSF);
}

// ---------------------------------------------------------------------------
// f32 WMMA GEMM, LDS-staged B:  C[M,N] = A[M,K] @ B[K,N],  N = NT*16.
// B rows are staged into LDS in K-tiles of 64 via CDNA5 async global->LDS
// loads; each wave owns one 16-row strip and issues NT WMMAs per A-fragment.
// M multiple of 16 preferred (ragged strips exit after cooperative phases);
// K multiple of 64.
template <int NT>
__global__ void k_gemm_wmma_lds(const float* __restrict__ A, const float* __restrict__ Bm,
                                float* __restrict__ C, int M, int K,
                                int lda, int ldbg, int ldc) {
    constexpr int N   = NT * 16;
    constexpr int LDB = N + 8;      // pad: lane halves of a B-frag hit disjoint banks
    constexpr int KT  = 64;         // K staging tile
    __shared__ float Bs[KT * LDB];

    int tid  = threadIdx.x;
    int wave = tid >> 5, lane = tid & 31;
    int strip = blockIdx.x * (blockDim.x >> 5) + wave;
    int l = lane & 15;              // A row / B,C column within tile
    int half2 = (lane >> 4) << 1;   // K sub-offset 0 or 2 per lane half
    bool active = (strip << 4) < M;
    const float* Arow = A + (size_t)((active ? (strip << 4) : 0) + l) * lda;

    v8f acc[NT] = {};
    for (int kk = 0; kk < K; kk += KT) {
        // cooperative async stage of B rows kk..kk+KT-1 into LDS (b128 chunks)
        for (int c = tid; c < KT * (N / 4); c += blockDim.x) {
            int row  = c / (N / 4);
            int col4 = (c - row * (N / 4)) * 4;
            const float* g = Bm + (size_t)(kk + row) * ldbg + col4;
            lds_float* p3 = (lds_float*)&Bs[row * LDB + col4];
            asm volatile("global_load_async_to_lds_b128 %0, %1, off"
                         :: "v"(p3), "v"(g) : "memory");
        }
        asm volatile("s_wait_asynccnt 0" ::: "memory");
        __syncthreads();

        if (active) {
            for (int k0 = 0; k0 < KT; k0 += 4) {
                int kb = k0 + half2;
                v2f a = *(const v2f*)(Arow + kk + kb);   // A[m][kb..kb+1]
#pragma unroll
                for (int t = 0; t < NT; ++t) {
                    v2f b;
                    b.x = Bs[kb * LDB + t * 16 + l];     // B[kb][n]   (ds_load)
                    b.y = Bs[(kb + 1) * LDB + t * 16 + l];
                    acc[t] = __builtin_amdgcn_wmma_f32_16x16x4_f32(
                        false, a, false, b, (short)0, acc[t], false, false);
                }
            }
        }
        __syncthreads();   // protect Bs before next stage
    }

    if (active) {
        int mb = (strip << 4) + ((lane >> 4) << 3);      // M base: +0 or +8
#pragma unroll
        for (int t = 0; t < NT; ++t) {
            float* Cp = C + (size_t)mb * ldc + t * 16 + l;
#pragma unroll
            for (int v = 0; v < 8; ++v) Cp[(size_t)v * ldc] = acc[t][v];
        }
    }
}

// ---------------------------------------------------------------------------
// rel-edge spmm restricted to batch users; acc slot = [sum item_emb | sum prop]
__global__ void k_rel_acc(const int* __restrict__ rows, const int* __restrict__ cols,
                          const int* __restrict__ u_slot, const float* __restrict__ ie,
                          const float* __restrict__ prop, float* __restrict__ acc) {
    int e = blockIdx.x * blockDim.x + threadIdx.x;
    if (e >= NNZ_E) return;
    int s = u_slot[rows[e]];
    if (s < 0) return;                              // ~99.5% early-out
    int c = cols[e];
    const float* iep = ie   + (size_t)c * EMBED;
    const float* pp  = prop + (size_t)c * EMBED;
    float* ap = acc + (size_t)s * 2 * EMBED;
    for (int j = 0; j < EMBED; ++j) {
        atomicAdd(ap + j,          iep[j]);
        atomicAdd(ap + EMBED + j,  pp[j]);
    }
}

// duplicate users: copy accumulation from the winning slot
__global__ void k_fixup(const int* __restrict__ user, const int* __restrict__ u_slot,
                        float* __restrict__ acc) {
    int idx = blockIdx.x * blockDim.x + threadIdx.x;
    if (idx >= BATCH * 2 * EMBED) return;
    int b = idx >> 7, j = idx & 127;
    int rep = u_slot[user[b]];
    if (rep != b) acc[idx] = acc[(size_t)rep * 2 * EMBED + j];
}

// normalize acc by ub_deg, accumulate user_feature += ubw * neigh
__global__ void k_norm_acc_uf(const int* __restrict__ user, const float* __restrict__ ub_deg,
                              const float* __restrict__ mw, float* __restrict__ acc,
                              float* __restrict__ ufeat, int r) {
    int idx = blockIdx.x * blockDim.x + threadIdx.x;
    if (idx >= BATCH * 2 * EMBED) return;
    int b = idx >> 7, j = idx & 127;
    int u = user[b];
    float deg = ub_deg[(size_t)r * NUM_USERS + u];
    float val = acc[idx] / (deg + EPSF);
    acc[idx] = val;
    if (j < EMBED) {
        float tw = 0.f;
#pragma unroll
        for (int q = 0; q < RREL; ++q) tw += ub_deg[(size_t)q * NUM_USERS + u] * mw[q];
        float w = deg * mw[r] / (tw + EPSF);
        ufeat[b * EMBED + j] += w * val;
    }
}

// score2 partial: out[b,k] += proj[b]·[item_emb[it] | prop[it]]
__global__ void k_score2(const int* __restrict__ item, const float* __restrict__ proj,
                         const float* __restrict__ ie, const float* __restrict__ prop,
                         float* __restrict__ out) {
    int idx = blockIdx.x * blockDim.x + threadIdx.x;
    if (idx >= BATCH * KITEMS) return;
    int b = idx / KITEMS;
    int it = item[idx];
    const float* pj = proj + (size_t)b * 2 * EMBED;
    const float* c0 = ie   + (size_t)it * EMBED;
    const float* c1 = prop + (size_t)it * EMBED;
    float s = 0.f;
    for (int j = 0; j < EMBED; ++j) s += pj[j] * c0[j] + pj[EMBED + j] * c1[j];
    out[idx] += s;
}

// item_feature[col] += user_embedding[row]  (64 lanes per train edge)
__global__ void k_itemfeat(const int* __restrict__ tr, const int* __restrict__ tc,
                           const float* __restrict__ ue, float* __restrict__ itf) {
    int idx = blockIdx.x * blockDim.x + threadIdx.x;
    if (idx >= NNZ_E * EMBED) return;
    int e = idx >> 6, j = idx & 63;
    atomicAdd(&itf[(size_t)tc[e] * EMBED + j], ue[(size_t)tr[e] * EMBED + j]);
}

// u_g[:, 0:64] = user_embedding[user]   (upper half filled by WMMA GEMM)
__global__ void k_ug_fill(const int* __restrict__ user, const float* __restrict__ ue,
                          float* __restrict__ u_g) {
    int idx = blockIdx.x * blockDim.x + threadIdx.x;
    if (idx >= BATCH * EMBED) return;
    int b = idx >> 6, j = idx & 63;
    u_g[(size_t)b * 2 * EMBED + j] = ue[(size_t)user[b] * EMBED + j];
}

// final: scores = score1 + (LAMB/R)*score2acc ; fused l2 reduction
__global__ void k_final(const int* __restrict__ item, const float* __restrict__ u_g,
                        const float* __restrict__ ie, const float* __restrict__ itfW,
                        float* __restrict__ out) {
    int idx = blockIdx.x * blockDim.x + threadIdx.x;
    if (idx >= BATCH * KITEMS) return;
    int b = idx / KITEMS, k = idx % KITEMS;
    int it = item[idx];
    const float* ug = u_g  + (size_t)b * 2 * EMBED;
    const float* c0 = ie   + (size_t)it * EMBED;
    const float* c1 = itfW + (size_t)it * EMBED;
    float s1 = 0.f, sq = 0.f;
    for (int j = 0; j < EMBED; ++j) {
        float a0 = ug[j], a1 = ug[EMBED + j];
        float e0 = c0[j], e1 = c1[j];
        s1 += a0 * e0 + a1 * e1;
        sq += e0 * e0 + e1 * e1;                     // i_g^2 contribution
        if (k == 0) sq += (float)KITEMS * (a0 * a0 + a1 * a1);  // K * u_g^2
    }
    out[idx] = s1 + (LAMBF / (float)RREL) * out[idx];
    // wave32 reduction, then one atomic per wave into the loss slot
    for (int off = 16; off > 0; off >>= 1) sq += __shfl_down(sq, off, 32);
    if ((threadIdx.x & 31) == 0) atomicAdd(out + BATCH * KITEMS, L2NORMF * sq);
}

// ---------------------------------------------------------------------------
static inline int nblk(long long n, int t) { return (int)((n + t - 1) / t); }
static inline int gemmblk(int M) { return (M / 16 + 7) / 8; }   // 8 waves/block

extern "C" void kernel_launch(void* const* d_in, const int* in_sizes, int n_in,
                              void* d_out, int out_size, void* d_ws, size_t ws_size,
                              hipStream_t stream) {
    const int*   user     = (const int*)  d_in[0];
    const int*   item     = (const int*)  d_in[1];
    const int*   rel_rows = (const int*)  d_in[2];
    const int*   rel_cols = (const int*)  d_in[3];
    const int*   ig_rows  = (const int*)  d_in[4];
    const int*   ig_cols  = (const int*)  d_in[5];
    const float* ig_vals  = (const float*)d_in[6];
    const int*   tr_rows  = (const int*)  d_in[7];
    const int*   tr_cols  = (const int*)  d_in[8];
    const float* ue       = (const float*)d_in[9];
    const float* ie       = (const float*)d_in[10];
    const float* mw       = (const float*)d_in[11];
    const float* ibW      = (const float*)d_in[12];   // (R,128,128)
    const float* ipW      = (const float*)d_in[13];   // (R,64,64)
    const float* W        = (const float*)d_in[14];   // (64,64)
    float* out = (float*)d_out;

    // workspace layout (floats)
    float* ws = (float*)d_ws;
    size_t o = 0;
    float* ub_deg = ws + o; o += (size_t)RREL * NUM_USERS;       // 300000
    int*   u_slot = (int*)(ws + o); o += NUM_USERS;              // 100000
    float* ig_deg = ws + o; o += NUM_ITEMS;                      // 50000
    float* agg    = ws + o; o += (size_t)NUM_ITEMS * EMBED;      // 3.2M (reused as itfW)
    float* prop   = ws + o; o += (size_t)NUM_ITEMS * EMBED;      // 3.2M
    float* acc    = ws + o; o += (size_t)BATCH * 2 * EMBED;      // 65536
    float* proj   = ws + o; o += (size_t)BATCH * 2 * EMBED;      // 65536
    float* ufeat  = ws + o; o += (size_t)BATCH * EMBED;          // 32768
    float* u_g    = ws + o; o += (size_t)BATCH * 2 * EMBED;      // 65536
    float* itf    = ws + o; o += (size_t)NUM_ITEMS * EMBED;      // 3.2M
    (void)ws_size; (void)n_in; (void)in_sizes;

    const int T = 256;

    // global zeroing (graph-capture safe memset nodes)
    hipMemsetAsync(ub_deg, 0,    (size_t)RREL * NUM_USERS * sizeof(float), stream);
    hipMemsetAsync(u_slot, 0xFF, (size_t)NUM_USERS * sizeof(int),          stream);
    hipMemsetAsync(ufeat,  0,    (size_t)BATCH * EMBED * sizeof(float),    stream);
    hipMemsetAsync(itf,    0,    (size_t)NUM_ITEMS * EMBED * sizeof(float),stream);
    hipMemsetAsync(d_out,  0,    (size_t)out_size * sizeof(float),         stream);

    k_slot<<<nblk(BATCH, T), T, 0, stream>>>(user, u_slot);
    k_ubdeg<<<nblk((long long)RREL * NNZ_E, T), T, 0, stream>>>(rel_rows, ub_deg);

    for (int r = 0; r < RREL; ++r) {
        const int*   igr = ig_rows  + (size_t)r * IG_NNZ_E;
        const int*   igc = ig_cols  + (size_t)r * IG_NNZ_E;
        const float* igv = ig_vals  + (size_t)r * IG_NNZ_E;
        const int*   rr  = rel_rows + (size_t)r * NNZ_E;
        const int*   rc  = rel_cols + (size_t)r * NNZ_E;

        hipMemsetAsync(ig_deg, 0, (size_t)NUM_ITEMS * sizeof(float),         stream);
        hipMemsetAsync(agg,    0, (size_t)NUM_ITEMS * EMBED * sizeof(float), stream);
        hipMemsetAsync(acc,    0, (size_t)BATCH * 2 * EMBED * sizeof(float), stream);

        k_igdeg<<<nblk(IG_NNZ_E, T), T, 0, stream>>>(igr, igv, ig_deg);
        k_agg<<<nblk((long long)IG_NNZ_E * EMBED, T), T, 0, stream>>>(igr, igc, igv, ie, agg);
        k_norm_agg<<<nblk((long long)NUM_ITEMS * EMBED, T), T, 0, stream>>>(ig_deg, agg);

        // prop = item_agg @ item_propagate_W[r]   (50000x64 @ 64x64) — WMMA + async-LDS
        k_gemm_wmma_lds<4><<<gemmblk(NUM_ITEMS), T, 0, stream>>>(
            agg, ipW + (size_t)r * EMBED * EMBED, prop,
            NUM_ITEMS, EMBED, EMBED, EMBED, EMBED);

        k_rel_acc<<<nblk(NNZ_E, T), T, 0, stream>>>(rr, rc, u_slot, ie, prop, acc);
        k_fixup<<<nblk(BATCH * 2 * EMBED, T), T, 0, stream>>>(user, u_slot, acc);
        k_norm_acc_uf<<<nblk(BATCH * 2 * EMBED, T), T, 0, stream>>>(user, ub_deg, mw, acc, ufeat, r);

        // proj = tmp_user_item_p @ item_behavior_W[r]   (512x128 @ 128x128) — WMMA + async-LDS
        k_gemm_wmma_lds<8><<<gemmblk(BATCH), T, 0, stream>>>(
            acc, ibW + (size_t)r * 4 * EMBED * EMBED, proj,
            BATCH, 2 * EMBED, 2 * EMBED, 2 * EMBED, 2 * EMBED);

        k_score2<<<nblk(BATCH * KITEMS, T), T, 0, stream>>>(item, proj, ie, prop, out);
    }

    // item_feature = segment_sum(user_embedding[train_rows], train_cols)
    k_itemfeat<<<nblk((long long)NNZ_E * EMBED, T), T, 0, stream>>>(tr_rows, tr_cols, ue, itf);

    // u_g = [user_embedding[user] | user_feature @ W] — WMMA into strided C
    k_ug_fill<<<nblk(BATCH * EMBED, T), T, 0, stream>>>(user, ue, u_g);
    k_gemm_wmma_lds<4><<<gemmblk(BATCH), T, 0, stream>>>(
        ufeat, W, u_g + EMBED, BATCH, EMBED, EMBED, EMBED, 2 * EMBED);

    // itfW = item_feature @ W  (50000x64 @ 64x64) — WMMA, reuse agg buffer
    k_gemm_wmma_lds<4><<<gemmblk(NUM_ITEMS), T, 0, stream>>>(
        itf, W, agg, NUM_ITEMS, EMBED, EMBED, EMBED, EMBED);

    // scores = score1 + (LAMB/R)*score2 ; fused l2 loss into out[B*K]
    k_final<<<nblk(BATCH * KITEMS, T), T, 0, stream>>>(item, u_g, ie, agg, out);
}